// RNN_77738908058364
// MI455X (gfx1250) — compile-verified
//
#include <hip/hip_runtime.h>
#include <math.h>

// Problem constants
#define T_LEN 2048
#define B_SZ  2
#define I_SZ  40
#define H_SZ  44
#define A_SZ  262144
#define G4    176          // 4*H
#define ROWS  (T_LEN*B_SZ) // 4096 pregate rows

typedef float v2f __attribute__((ext_vector_type(2)));
typedef float v8f __attribute__((ext_vector_type(8)));

__device__ __forceinline__ float sigf(float x) { return 1.0f / (1.0f + __expf(-x)); }

// ---------------------------------------------------------------------------
// Kernel 1: pregates P[row=t*2+b][n] = x[row] @ W_ih^T + (b_ih + b_hh)
// grid = 256 M-tiles of 16 rows, block = 11 waves (352 threads), one N-tile/wave
// ---------------------------------------------------------------------------
__global__ void k_pregate(const float* __restrict__ x,
                          const float* __restrict__ Wih,
                          const float* __restrict__ bih,
                          const float* __restrict__ bhh,
                          float* __restrict__ P) {
  const int wave = threadIdx.x >> 5;
  const int lane = threadIdx.x & 31;
  const int m    = lane & 15;               // A-row within tile
  const int kk   = (lane >> 4) * 2;         // K sub-offset (0 or 2)
  const int n    = wave * 16 + (lane & 15); // output column 0..175
  const int row0 = blockIdx.x * 16;

  const float bias = bih[n] + bhh[n];
  v8f c;
  #pragma unroll
  for (int v = 0; v < 8; ++v) c[v] = bias;  // every valid row gets the bias

  const float* xr = x   + (size_t)(row0 + m) * I_SZ + kk;
  const float* wr = Wih + (size_t)n * I_SZ + kk;
  #pragma unroll
  for (int k = 0; k < I_SZ; k += 4) {
    v2f a, b;
    a.x = xr[k];  a.y = xr[k + 1];
    b.x = wr[k];  b.y = wr[k + 1];
    c = __builtin_amdgcn_wmma_f32_16x16x4_f32(false, a, false, b, (short)0, c,
                                              false, false);
  }

  const int mrow = (lane >> 4) * 8;         // C/D row base for this half-wave
  #pragma unroll
  for (int v = 0; v < 8; ++v)
    P[(size_t)(row0 + mrow + v) * G4 + n] = c[v];
}

// ---------------------------------------------------------------------------
// Kernel 2: sequential LSTM scan, single workgroup (11 waves).
// Each step: g = pregate[t] + h @ W_hh^T via TWO interleaved WMMA chains
// (even/odd K-chunks) to halve the serial RAW latency. W_hh fragments are
// loop-invariant and live in registers; h/c live in LDS; next step's pregates
// are prefetched under the activation phase.
// Finishes by writing base[b] = [h;c] . w_e + b_v.
// ---------------------------------------------------------------------------
__global__ void k_lstm(const float* __restrict__ P,
                       const float* __restrict__ Whh,
                       const float* __restrict__ h0,
                       const float* __restrict__ c0,
                       const float* __restrict__ Wv,
                       const float* __restrict__ bv,
                       float* __restrict__ base) {
  __shared__ float hl[B_SZ * H_SZ];   // h state
  __shared__ float cl[B_SZ * H_SZ];   // c state
  __shared__ float gl[B_SZ * G4];     // gates of current step

  const int tid  = threadIdx.x;
  const int wave = tid >> 5;
  const int lane = tid & 31;
  const int m    = lane & 15;
  const int kk   = (lane >> 4) * 2;
  const int n    = wave * 16 + (lane & 15);

  // Preload this wave's W_hh^T B-fragments (loop invariant): K=44 -> 11 chunks
  v2f bw[11];
  #pragma unroll
  for (int k = 0; k < 11; ++k) {
    const float* wr = Whh + (size_t)n * H_SZ + 4 * k + kk;
    bw[k].x = wr[0];
    bw[k].y = wr[1];
  }

  if (tid < B_SZ * H_SZ) {            // init state
    hl[tid] = h0[tid];
    cl[tid] = c0[tid];
  }
  __syncthreads();

  // A-row clamp: rows 2..15 of the A fragment only affect D rows we never
  // read, so feed them copies of rows 0/1 -> no EXEC-divergent selects.
  const int arow = (m & 1) * H_SZ;

  // Prefetch pregates for t = 0 (upper lanes load duplicates; their C rows
  // are ignored anyway).
  float p0 = P[n];
  float p1 = P[G4 + n];

  for (int t = 0; t < T_LEN; ++t) {
    v8f ca = {};
    ca[0] = p0;                       // row 0 (batch 0) init
    ca[1] = p1;                       // row 1 (batch 1) init
    v8f cb = {};

    // Two interleaved WMMA chains over the 11 K-chunks
    #pragma unroll
    for (int k = 0; k < 11; ++k) {
      const v2f a = *(const v2f*)(hl + arow + 4 * k + kk);  // 8B LDS load
      if (k & 1)
        cb = __builtin_amdgcn_wmma_f32_16x16x4_f32(false, a, false, bw[k],
                                                   (short)0, cb, false, false);
      else
        ca = __builtin_amdgcn_wmma_f32_16x16x4_f32(false, a, false, bw[k],
                                                   (short)0, ca, false, false);
    }

    // Prefetch next step's pregates (latency hidden under barriers/activations)
    if (t + 1 < T_LEN) {
      p0 = P[(size_t)(2 * t + 2) * G4 + n];
      p1 = P[(size_t)(2 * t + 3) * G4 + n];
    }

    const float g0 = ca[0] + cb[0];
    const float g1 = ca[1] + cb[1];
    if (lane < 16) {                  // extract gate rows 0,1
      gl[0 * G4 + n] = g0;
      gl[1 * G4 + n] = g1;
    }
    __syncthreads();

    if (tid < B_SZ * H_SZ) {          // cell update: 88 lanes
      const int b = tid / H_SZ, j = tid % H_SZ;
      const float gi = gl[b * G4 + j];
      const float gf = gl[b * G4 + H_SZ + j];
      const float gg = gl[b * G4 + 2 * H_SZ + j];
      const float go = gl[b * G4 + 3 * H_SZ + j];
      const float cn = sigf(gf) * cl[tid] + sigf(gi) * tanhf(gg);
      cl[tid] = cn;
      hl[tid] = sigf(go) * tanhf(cn);
    }
    __syncthreads();
  }

  // base[b] = sum_j h[b][j]*Wv[j] + c[b][j]*Wv[44+j] + b_v
  if (tid < B_SZ) {
    float s = bv[0];
    #pragma unroll 4
    for (int j = 0; j < H_SZ; ++j) {
      s += hl[tid * H_SZ + j] * Wv[j];
      s += cl[tid * H_SZ + j] * Wv[H_SZ + j];
    }
    base[tid] = s;
  }
  (void)wave;
}

// ---------------------------------------------------------------------------
// Kernel 3: streaming value head (bandwidth bound: 84 MB of action).
// One thread per (b,a) row: 10x global_load_b128 + scalar-operand FMAs.
// ---------------------------------------------------------------------------
__global__ void k_value(const float* __restrict__ action,
                        const int* __restrict__ act_len,
                        const float* __restrict__ Wv,
                        const float* __restrict__ base,
                        float* __restrict__ out) {
  const size_t idx = (size_t)blockIdx.x * blockDim.x + threadIdx.x; // 0..B*A-1
  const int b = (int)(idx >> 18);          // A = 2^18
  const int a = (int)(idx & (A_SZ - 1));

  const float* __restrict__ w   = Wv + 2 * H_SZ;   // w_a (uniform -> SGPRs)
  const float* __restrict__ row = action + idx * (size_t)I_SZ;

  float s = 0.0f;
  #pragma unroll
  for (int k = 0; k < I_SZ; k += 4) {
    const float4 v = *(const float4*)(row + k);    // 160B rows, 16B aligned
    s += v.x * w[k] + v.y * w[k + 1] + v.z * w[k + 2] + v.w * w[k + 3];
  }
  const float val = base[b] + s;
  out[idx] = (a < act_len[b]) ? val : 0.0f;
}

// ---------------------------------------------------------------------------
extern "C" void kernel_launch(void* const* d_in, const int* in_sizes, int n_in,
                              void* d_out, int out_size, void* d_ws, size_t ws_size,
                              hipStream_t stream) {
  (void)in_sizes; (void)n_in; (void)out_size; (void)ws_size;

  const float* x      = (const float*)d_in[0];
  const float* h0     = (const float*)d_in[1];
  const float* c0     = (const float*)d_in[2];
  const float* action = (const float*)d_in[3];
  const int*   alen   = (const int*)d_in[4];
  const float* Wih    = (const float*)d_in[5];
  const float* Whh    = (const float*)d_in[6];
  const float* bih    = (const float*)d_in[7];
  const float* bhh    = (const float*)d_in[8];
  const float* Wv     = (const float*)d_in[9];
  const float* bv     = (const float*)d_in[10];
  float* out = (float*)d_out;

  float* P    = (float*)d_ws;            // 4096 x 176 pregates (2.88 MB)
  float* base = P + (size_t)ROWS * G4;   // 2 floats

  k_pregate<<<ROWS / 16, 352, 0, stream>>>(x, Wih, bih, bhh, P);
  k_lstm<<<1, 352, 0, stream>>>(P, Whh, h0, c0, Wv, bv, base);
  k_value<<<(B_SZ * A_SZ) / 256, 256, 0, stream>>>(action, alen, Wv, base, out);
}